// FPS_VoxelsEmbedding_5592047420188
// MI455X (gfx1250) — compile-verified
//
#include <hip/hip_runtime.h>
#include <hip/hip_bf16.h>

typedef __attribute__((ext_vector_type(2))) float v2f;
typedef __attribute__((ext_vector_type(8))) float v8f;

#define B_    16
#define N_    4096
#define C_    64
#define CIN   67          // 3 + C
#define CP    68          // padded K dim (multiple of 4)
#define OUT_  128
#define S_    1024
#define K_    32

// ---------------------------------------------------------------------------
// Kernel 1: farthest point sampling. One block per batch. xyz in LDS,
// running min-distances in registers, argmax via wave32 shuffle reduction.
// Writes new_xyz (tail segment of d_out).
// ---------------------------------------------------------------------------
__global__ __launch_bounds__(256)
void fps_kernel(const float* __restrict__ xyz, float* __restrict__ newxyz)
{
    __shared__ float xs[N_], ys[N_], zs[N_];
    __shared__ float redv[8];
    __shared__ int   redi[8];
    __shared__ int   sFar;

    const int b    = blockIdx.x;
    const int t    = threadIdx.x;
    const int lane = t & 31;
    const int wid  = t >> 5;
    const float* base = xyz + (size_t)b * N_ * 3;

    for (int i = t; i < N_; i += 256) {
        xs[i] = base[i * 3 + 0];
        ys[i] = base[i * 3 + 1];
        zs[i] = base[i * 3 + 2];
    }
    __syncthreads();

    float dist[16];
#pragma unroll
    for (int j = 0; j < 16; ++j) dist[j] = 1e10f;

    int far = 0;
    for (int s = 0; s < S_; ++s) {
        const float cx = xs[far], cy = ys[far], cz = zs[far];
        if (t == 0) {
            float* o = newxyz + ((size_t)b * S_ + s) * 3;
            o[0] = cx; o[1] = cy; o[2] = cz;
        }
        // update running min distance; track local argmax (lowest index on tie)
        float bv = -1.0f; int bi = 0;
#pragma unroll
        for (int j = 0; j < 16; ++j) {
            const int i = t + j * 256;
            const float dx = xs[i] - cx, dy = ys[i] - cy, dz = zs[i] - cz;
            const float d  = dx * dx + dy * dy + dz * dz;
            const float nd = fminf(dist[j], d);
            dist[j] = nd;
            if (nd > bv || (nd == bv && i < bi)) { bv = nd; bi = i; }
        }
#pragma unroll
        for (int off = 16; off >= 1; off >>= 1) {
            const float ov = __shfl_xor(bv, off, 32);
            const int   oi = __shfl_xor(bi, off, 32);
            if (ov > bv || (ov == bv && oi < bi)) { bv = ov; bi = oi; }
        }
        if (lane == 0) { redv[wid] = bv; redi[wid] = bi; }
        __syncthreads();
        if (wid == 0) {
            float v = (lane < 8) ? redv[lane] : -1.0f;
            int   i = (lane < 8) ? redi[lane] : 0x7fffffff;
#pragma unroll
            for (int off = 4; off >= 1; off >>= 1) {
                const float ov = __shfl_xor(v, off, 32);
                const int   oi = __shfl_xor(i, off, 32);
                if (ov > v || (ov == v && oi < i)) { v = ov; i = oi; }
            }
            if (lane == 0) sFar = i;
        }
        __syncthreads();
        far = sFar;
    }
}

// ---------------------------------------------------------------------------
// Kernel 2: kNN (top-32 smallest squared distances). One block per query.
// Distances cached once in LDS (16KB), then 32 argmin+exclude passes.
// ---------------------------------------------------------------------------
__global__ __launch_bounds__(256)
void knn_kernel(const float* __restrict__ xyz, const float* __restrict__ newxyz,
                int* __restrict__ knn)
{
    __shared__ float dist[N_];
    __shared__ float redv[8];
    __shared__ int   redi[8];

    const int gq   = blockIdx.x;          // b*S + s
    const int b    = gq >> 10;
    const int t    = threadIdx.x;
    const int lane = t & 31;
    const int wid  = t >> 5;

    const float cx = newxyz[gq * 3 + 0];
    const float cy = newxyz[gq * 3 + 1];
    const float cz = newxyz[gq * 3 + 2];
    const float* base = xyz + (size_t)b * N_ * 3;

#pragma unroll
    for (int j = 0; j < 16; ++j) {
        const int i = t + j * 256;
        const float dx = base[i * 3 + 0] - cx;
        const float dy = base[i * 3 + 1] - cy;
        const float dz = base[i * 3 + 2] - cz;
        dist[i] = dx * dx + dy * dy + dz * dz;
    }
    __syncthreads();

    for (int k = 0; k < K_; ++k) {
        float bv = 3.4e38f; int bi = 0x7fffffff;
#pragma unroll
        for (int j = 0; j < 16; ++j) {
            const int i = t + j * 256;
            const float v = dist[i];
            if (v < bv || (v == bv && i < bi)) { bv = v; bi = i; }
        }
#pragma unroll
        for (int off = 16; off >= 1; off >>= 1) {
            const float ov = __shfl_xor(bv, off, 32);
            const int   oi = __shfl_xor(bi, off, 32);
            if (ov < bv || (ov == bv && oi < bi)) { bv = ov; bi = oi; }
        }
        if (lane == 0) { redv[wid] = bv; redi[wid] = bi; }
        __syncthreads();
        if (t == 0) {
            float v = redv[0]; int i = redi[0];
#pragma unroll
            for (int w = 1; w < 8; ++w)
                if (redv[w] < v || (redv[w] == v && redi[w] < i)) { v = redv[w]; i = redi[w]; }
            knn[(size_t)gq * K_ + k] = i;
            dist[i] = 3.0e38f;           // exclude for next pass
        }
        __syncthreads();
    }
}

// ---------------------------------------------------------------------------
// Kernel 3: gather + (32x68)x(68x128) f32 WMMA GEMM + max-over-K + LayerNorm.
// 8 waves/block; wave w owns output columns [16w,16w+16). 8 queries per block.
//
// f32 16x16x4 fragment layout (ISA 7.12.2):
//   A[16x4]: lane l holds A[l&15][2*(l>>4)+{0,1}]  -> float2 from LDS
//   B[4x16]: lane l holds B[2*(l>>4)+{0,1}][l&15]  = W[(l&15)+16w][k..k+1]
//   C[16x16]: VGPR v = rows {v, v+8}, cols = lane&15  -> max over rows is
//             8 v_max + one shfl_xor(16).
// ---------------------------------------------------------------------------
__global__ __launch_bounds__(256)
void embed_kernel(const float* __restrict__ xyz, const float* __restrict__ voxels,
                  const float* __restrict__ conv_w, const float* __restrict__ conv_b,
                  const float* __restrict__ gamma, const float* __restrict__ beta,
                  const float* __restrict__ newxyz, const int* __restrict__ knn,
                  float* __restrict__ out)
{
    __shared__ float Wlds[OUT_ * CP];   // 34816 B, W[n][c], c padded to 68
    __shared__ float blds[OUT_];
    __shared__ float feat[K_ * CP];     // 8704 B
    __shared__ float outv[OUT_];
    __shared__ float stats[2];

    const int t    = threadIdx.x;
    const int lane = t & 31;
    const int w    = t >> 5;

    for (int i = t; i < OUT_ * CP; i += 256) {
        const int n = i / CP, c = i % CP;
        Wlds[i] = (c < CIN) ? conv_w[n * CIN + c] : 0.0f;
    }
    if (t < OUT_) blds[t] = conv_b[t];
    __syncthreads();

    for (int ql = 0; ql < 8; ++ql) {
        const int gq = blockIdx.x * 8 + ql;
        const int b  = gq >> 10;
        const float cx = newxyz[gq * 3 + 0];
        const float cy = newxyz[gq * 3 + 1];
        const float cz = newxyz[gq * 3 + 2];

        // gather: 8 threads per neighbor, stripe over channels
        {
            const int kn  = t >> 3;
            const int sub = t & 7;
            const int idx = knn[(size_t)gq * K_ + kn];
            const float* xp = xyz    + ((size_t)b * N_ + idx) * 3;
            const float* vp = voxels + ((size_t)b * N_ + idx) * C_;
            for (int c = sub; c < CP; c += 8) {
                float val;
                if      (c == 0)  val = xp[0] - cx;
                else if (c == 1)  val = xp[1] - cy;
                else if (c == 2)  val = xp[2] - cz;
                else if (c < CIN) val = vp[c - 3];
                else              val = 0.0f;
                feat[kn * CP + c] = val;
            }
        }
        __syncthreads();

        // GEMM: acc[32x16 cols of wave w] += feat * W^T
        v8f acc0 = {}; v8f acc1 = {};
        const int m  = lane & 15;
        const int kp = lane >> 4;
#pragma unroll
        for (int ks = 0; ks < 17; ++ks) {
            const int kb = ks * 4 + kp * 2;
            const v2f a0 = *(const v2f*)(&feat[m * CP + kb]);
            const v2f a1 = *(const v2f*)(&feat[(m + 16) * CP + kb]);
            const v2f bf = *(const v2f*)(&Wlds[(w * 16 + m) * CP + kb]);
            acc0 = __builtin_amdgcn_wmma_f32_16x16x4_f32(
                       false, a0, false, bf, (short)0, acc0, false, false);
            acc1 = __builtin_amdgcn_wmma_f32_16x16x4_f32(
                       false, a1, false, bf, (short)0, acc1, false, false);
        }

        // max over the 32 neighbor rows, then +bias
        float mv = fmaxf(acc0[0], acc1[0]);
#pragma unroll
        for (int v = 1; v < 8; ++v) mv = fmaxf(mv, fmaxf(acc0[v], acc1[v]));
        mv = fmaxf(mv, __shfl_xor(mv, 16, 32));
        if (lane < 16) outv[w * 16 + lane] = mv + blds[w * 16 + lane];
        __syncthreads();

        // LayerNorm over 128 channels (wave 0 computes stats)
        if (w == 0) {
            float s1 = 0.0f, s2 = 0.0f;
#pragma unroll
            for (int j = 0; j < 4; ++j) {
                const float x = outv[lane * 4 + j];
                s1 += x; s2 += x * x;
            }
#pragma unroll
            for (int off = 16; off >= 1; off >>= 1) {
                s1 += __shfl_xor(s1, off, 32);
                s2 += __shfl_xor(s2, off, 32);
            }
            if (lane == 0) {
                const float mu  = s1 * (1.0f / OUT_);
                const float var = s2 * (1.0f / OUT_) - mu * mu;
                stats[0] = mu;
                stats[1] = rsqrtf(var + 1e-5f);
            }
        }
        __syncthreads();
        if (t < OUT_) {
            const float o = (outv[t] - stats[0]) * stats[1] * gamma[t] + beta[t];
            out[(size_t)gq * OUT_ + t] = o;
        }
        __syncthreads();   // protect feat/outv before next query
    }
}

// ---------------------------------------------------------------------------
extern "C" void kernel_launch(void* const* d_in, const int* in_sizes, int n_in,
                              void* d_out, int out_size, void* d_ws, size_t ws_size,
                              hipStream_t stream)
{
    const float* xyz    = (const float*)d_in[0];   // [16,4096,3]
    const float* voxels = (const float*)d_in[1];   // [16,4096,64]
    const float* conv_w = (const float*)d_in[2];   // [128,67]
    const float* conv_b = (const float*)d_in[3];   // [128]
    const float* gamma  = (const float*)d_in[4];   // [128]
    const float* beta   = (const float*)d_in[5];   // [128]

    float* out    = (float*)d_out;                         // [16,1024,128]
    float* newxyz = out + (size_t)B_ * S_ * OUT_;          // [16,1024,3]
    int*   knn    = (int*)d_ws;                            // [16,1024,32]

    fps_kernel  <<<B_,          256, 0, stream>>>(xyz, newxyz);
    knn_kernel  <<<B_ * S_,     256, 0, stream>>>(xyz, newxyz, knn);
    embed_kernel<<<(B_ * S_)/8, 256, 0, stream>>>(xyz, voxels, conv_w, conv_b,
                                                  gamma, beta, newxyz, knn, out);
}